// BlobbyGyroid_63196148793935
// MI455X (gfx1250) — compile-verified
//
#include <hip/hip_runtime.h>
#include <hip/hip_bf16.h>

#define BLK 256
#define LSTRIDE 21   // 21 coprime with 64 banks -> conflict-free row reads

typedef float v2f __attribute__((ext_vector_type(2)));
typedef float v8f __attribute__((ext_vector_type(8)));

struct BGParams {
  const float *p, *t;
  const float *warp_a, *warp_B, *warp_omega, *warp_phi;
  const float *omega, *phi, *alpha;
  const float *mb_w, *mb_beta_raw, *mb_cbar, *mb_u, *mb_nu, *mb_psi;
  const float *h_s, *h_k, *h_w, *h_zeta;
  const float *kappa, *bias_b, *delta_raw;
  const float *q0, *q1, *Q2, *q3, *eta, *w_vec, *zeta, *light_dir;
  float* out;
  int P;
};

__device__ __forceinline__ float softplusf(float x) {
  // numerically stable log1p(exp(x))
  return fmaxf(x, 0.f) + log1pf(__expf(-fabsf(x)));
}

__device__ __forceinline__ float fsigmoid(float x) {
  return __builtin_amdgcn_rcpf(1.f + __expf(-x));
}

// Per-wave dense contraction:  D[16x16] = A[16x4] * B[4x16] + C(bias per col),
// done for both 16-point tiles of the wave.  Each lane owns one point; its
// (x0..x3) row is scattered into the documented WMMA A layout via shuffles
// (lanes 0-15: K0/K1, lanes 16-31: K2/K3).  Results are transposed back
// through per-wave LDS so each lane receives its own point's 12 columns.
__device__ __forceinline__ void wave_mat4x16(
    float x0, float x1, float x2, float x3,
    float b0, float b1, float cbias,
    float* lds, int lane, float outRow[12])
{
  const int half = lane >> 4;
  const int l15  = lane & 15;
  v2f B; B[0] = b0; B[1] = b1;
  v8f C;
#pragma unroll
  for (int r = 0; r < 8; ++r) C[r] = cbias;
#pragma unroll
  for (int T = 0; T < 2; ++T) {
    const int src = T * 16 + l15;
    const float t0 = __shfl(x0, src, 32);
    const float t1 = __shfl(x1, src, 32);
    const float t2 = __shfl(x2, src, 32);
    const float t3 = __shfl(x3, src, 32);
    v2f A; A[0] = half ? t2 : t0; A[1] = half ? t3 : t1;
    // 8 args: (neg_a, A, neg_b, B, c_mod, C, reuse_a, reuse_b)
    v8f D = __builtin_amdgcn_wmma_f32_16x16x4_f32(
        false, A, false, B, (short)0, C, false, false);
    // C/D layout: lanes 0-15 -> N=lane, M=r ; lanes 16-31 -> N=lane-16, M=8+r
    const int rowBase = T * 16 + half * 8;
#pragma unroll
    for (int r = 0; r < 8; ++r)
      lds[(rowBase + r) * LSTRIDE + l15] = D[r];
  }
  __builtin_amdgcn_wave_barrier();   // same-wave LDS is in-order; just fence the scheduler
#pragma unroll
  for (int c = 0; c < 12; ++c) outRow[c] = lds[lane * LSTRIDE + c];
  __builtin_amdgcn_wave_barrier();   // buffer is reused by the next matmul
}

__global__ __launch_bounds__(BLK) void blobby_gyroid_kernel(BGParams prm)
{
  __shared__ float sBeta[16];
  __shared__ float sInvDelta;
  __shared__ float argLds[(BLK / 32) * 32 * LSTRIDE];

  const int tid  = threadIdx.x;
  const int lane = tid & 31;
  const int wave = tid >> 5;
  float* lds = argLds + wave * 32 * LSTRIDE;

  if (tid < 16) sBeta[tid] = softplusf(prm.mb_beta_raw[tid]) + 0.3f;
  if (tid == 16) sInvDelta = __builtin_amdgcn_rcpf(softplusf(prm.delta_raw[0]) + 0.001f);
  __syncthreads();

  int i = blockIdx.x * BLK + tid;
  if (i >= prm.P) i = prm.P - 1;        // keep EXEC all-ones for WMMA (P is a multiple of 256)

  const float px = prm.p[3*i+0], py = prm.p[3*i+1], pz = prm.p[3*i+2];
  const float tt = prm.t[i];

  // ----- per-lane B/C operands (lane -> output column) -----
  const int n = lane & 15; const int half = lane >> 4;

  // matmul 1: col (j,e): arg = B_j[e,:].p + omega_j*t + phi_j[e]
  float w1b0 = 0.f, w1b1 = 0.f, w1c = 0.f;
  if (n < 12) {
    const int j = n / 3, e = n % 3;
    const float* Bj = prm.warp_B + j*9 + e*3;
    if (!half) { w1b0 = Bj[0]; w1b1 = Bj[1]; }
    else       { w1b0 = Bj[2]; w1b1 = prm.warp_omega[j]; }
    w1c = prm.warp_phi[j*3 + e];
  }
  // matmul 2: cols 0-7 harmonic phases, col 8 emissive phase, cols 9-11 gyroid arg
  float w2b0 = 0.f, w2b1 = 0.f, w2c = 0.f;
  if (n < 8) {
    const float* hk = prm.h_k + n*3;
    if (!half) { w2b0 = hk[0]; w2b1 = hk[1]; }
    else       { w2b0 = hk[2]; w2b1 = prm.h_w[n]; }
    w2c = prm.h_zeta[n];
  } else if (n == 8) {
    const float et = prm.eta[0];
    if (!half) { w2b0 = et*prm.w_vec[0]; w2b1 = et*prm.w_vec[1]; }
    else       { w2b0 = et*prm.w_vec[2]; w2b1 = prm.zeta[0]; }
  } else if (n < 12) {
    const int d = n - 9;
    if (!half) { w2b0 = (d==0)?prm.omega[0]:0.f; w2b1 = (d==1)?prm.omega[1]:0.f; }
    else       { w2b0 = (d==2)?prm.omega[2]:0.f; w2b1 = 0.f; }
    w2c = prm.phi[d];
  }

  // ----- warp field (WMMA #1) -----
  float arg[12];
  wave_mat4x16(px, py, pz, tt, w1b0, w1b1, w1c, lds, lane, arg);

  float sj[12], cj[12];
  float qv[3] = {px, py, pz};
#pragma unroll
  for (int j = 0; j < 4; ++j)
#pragma unroll
    for (int e = 0; e < 3; ++e) {
      const int m = j*3 + e;
      sj[m] = __sinf(arg[m]);
      cj[m] = __cosf(arg[m]);
      qv[e] = fmaf(prm.warp_a[m], sj[m], qv[e]);
    }

  // ----- fused linear maps on (q,t) (WMMA #2) -----
  float lin[12];
  wave_mat4x16(qv[0], qv[1], qv[2], tt, w2b0, w2b1, w2c, lds, lane, lin);

  // gyroid + its gradient (lin[9..11] = omega*q + phi)
  float gs[3], gc[3];
#pragma unroll
  for (int d = 0; d < 3; ++d) { gs[d] = __sinf(lin[9+d]); gc[d] = __cosf(lin[9+d]); }
  const float G = gs[0]*gc[1] + gs[1]*gc[2] + gs[2]*gc[0] - prm.alpha[0];
  float gG[3];
  gG[0] = prm.omega[0] * (gc[0]*gc[1] - gs[2]*gs[0]);
  gG[1] = prm.omega[1] * (gc[1]*gc[2] - gs[0]*gs[1]);
  gG[2] = prm.omega[2] * (gc[2]*gc[0] - gs[1]*gs[2]);

  // metaballs + gradient
  float M = 0.f, gm[3] = {0.f, 0.f, 0.f};
#pragma unroll
  for (int b = 0; b < 16; ++b) {
    const float sn = __sinf(fmaf(prm.mb_nu[b], tt, prm.mb_psi[b]));
    const float dx = qv[0] - fmaf(prm.mb_u[b*3+0], sn, prm.mb_cbar[b*3+0]);
    const float dy = qv[1] - fmaf(prm.mb_u[b*3+1], sn, prm.mb_cbar[b*3+1]);
    const float dz = qv[2] - fmaf(prm.mb_u[b*3+2], sn, prm.mb_cbar[b*3+2]);
    const float d2 = fmaf(dx, dx, fmaf(dy, dy, dz*dz));
    const float beta = sBeta[b];
    const float we = prm.mb_w[b] * __expf(-beta * d2);
    M += we;
    const float g = -2.f * beta * we;
    gm[0] = fmaf(g, dx, gm[0]); gm[1] = fmaf(g, dy, gm[1]); gm[2] = fmaf(g, dz, gm[2]);
  }

  // harmonics + gradient (lin[0..7] are the phases)
  float H = 0.f, gH[3] = {0.f, 0.f, 0.f};
#pragma unroll
  for (int k = 0; k < 8; ++k) {
    const float hs = prm.h_s[k];
    H = fmaf(hs, __sinf(lin[k]), H);
    const float hc = hs * __cosf(lin[k]);
    gH[0] = fmaf(hc, prm.h_k[k*3+0], gH[0]);
    gH[1] = fmaf(hc, prm.h_k[k*3+1], gH[1]);
    gH[2] = fmaf(hc, prm.h_k[k*3+2], gH[2]);
  }

  const float kap = prm.kappa[0];
  const float F   = 0.1f*G + kap*log1pf(M) + H - prm.bias_b[0];
  const float cM  = kap * __builtin_amdgcn_rcpf(1.f + M);

  float dFdq[3];
#pragma unroll
  for (int d = 0; d < 3; ++d) dFdq[d] = 0.1f*gG[d] + cM*gm[d] + gH[d];

  // chain rule through the warp: grad_p = dFdq + B^T (a .* cos(arg) .* dFdq)
  float gp[3] = {dFdq[0], dFdq[1], dFdq[2]};
#pragma unroll
  for (int j = 0; j < 4; ++j)
#pragma unroll
    for (int e = 0; e < 3; ++e) {
      const int m = j*3 + e;
      const float wn = dFdq[e] * prm.warp_a[m] * cj[m];
      const float* Bj = prm.warp_B + j*9 + e*3;
      gp[0] = fmaf(wn, Bj[0], gp[0]);
      gp[1] = fmaf(wn, Bj[1], gp[1]);
      gp[2] = fmaf(wn, Bj[2], gp[2]);
    }

  // shading
  const float len2 = fmaf(gp[0], gp[0], fmaf(gp[1], gp[1], gp[2]*gp[2]));
  const float inv  = __builtin_amdgcn_rsqf(fmaxf(len2, 1e-24f)); // == 1/max(|g|,1e-12)
  const float nv0 = gp[0]*inv, nv1 = gp[1]*inv, nv2 = gp[2]*inv;

  const float sig   = softplusf(-F * sInvDelta);
  const float ndl   = nv0*prm.light_dir[0] + nv1*prm.light_dir[1] + nv2*prm.light_dir[2];
  const float emis  = __sinf(lin[8]);
  const float boost = fminf(fmaxf(sig * 5.f, 0.f), 1.f);

  const int P = prm.P;
  float* out = prm.out;
  out[i]         = sig;   // sigma [P,1]
  out[4*P + i]   = F;     // Fval  [P]
  const float nvv[3] = {nv0, nv1, nv2};
#pragma unroll
  for (int d = 0; d < 3; ++d) {
    const float x = prm.q0[d] + prm.q1[0]*ndl
                  + prm.Q2[d*3+0]*nv0 + prm.Q2[d*3+1]*nv1 + prm.Q2[d*3+2]*nv2
                  + prm.q3[0]*emis + 0.3f*__sinf(2.f*qv[d]) + 0.5f*boost;
    out[P   + 3*i + d] = fsigmoid(x);  // rgb [P,3]
    out[5*P + 3*i + d] = nvv[d];       // n   [P,3]
  }
}

extern "C" void kernel_launch(void* const* d_in, const int* in_sizes, int n_in,
                              void* d_out, int out_size, void* d_ws, size_t ws_size,
                              hipStream_t stream) {
  (void)n_in; (void)out_size; (void)d_ws; (void)ws_size;
  BGParams prm;
  prm.p           = (const float*)d_in[0];
  prm.t           = (const float*)d_in[1];
  prm.warp_a      = (const float*)d_in[2];
  prm.warp_B      = (const float*)d_in[3];
  prm.warp_omega  = (const float*)d_in[4];
  prm.warp_phi    = (const float*)d_in[5];
  prm.omega       = (const float*)d_in[6];
  prm.phi         = (const float*)d_in[7];
  prm.alpha       = (const float*)d_in[8];
  prm.mb_w        = (const float*)d_in[9];
  prm.mb_beta_raw = (const float*)d_in[10];
  prm.mb_cbar     = (const float*)d_in[11];
  prm.mb_u        = (const float*)d_in[12];
  prm.mb_nu       = (const float*)d_in[13];
  prm.mb_psi      = (const float*)d_in[14];
  prm.h_s         = (const float*)d_in[15];
  prm.h_k         = (const float*)d_in[16];
  prm.h_w         = (const float*)d_in[17];
  prm.h_zeta      = (const float*)d_in[18];
  prm.kappa       = (const float*)d_in[19];
  prm.bias_b      = (const float*)d_in[20];
  prm.delta_raw   = (const float*)d_in[21];
  prm.q0          = (const float*)d_in[22];
  prm.q1          = (const float*)d_in[23];
  prm.Q2          = (const float*)d_in[24];
  prm.q3          = (const float*)d_in[25];
  prm.eta         = (const float*)d_in[26];
  prm.w_vec       = (const float*)d_in[27];
  prm.zeta        = (const float*)d_in[28];
  prm.light_dir   = (const float*)d_in[29];
  prm.out         = (float*)d_out;
  prm.P           = in_sizes[1];  // t has P elements

  const int grid = (prm.P + BLK - 1) / BLK;
  blobby_gyroid_kernel<<<grid, BLK, 0, stream>>>(prm);
}